// PGNN_90220083020049
// MI455X (gfx1250) — compile-verified
//
#include <hip/hip_runtime.h>
#include <math.h>

typedef __attribute__((ext_vector_type(2))) float v2f;
typedef __attribute__((ext_vector_type(8))) float v8f;

// ---------------------------------------------------------------------------
// C[M,Nd] = A[M,Kd] @ W[Nd,Kd]^T + bias[Nd]
// One wave32 computes one 16x16 output tile via V_WMMA_F32_16X16X4_F32.
// A-frag (16x4 f32): lanes 0-15 hold K=kb+0,kb+1 ; lanes 16-31 hold kb+2,kb+3
// B-frag (4x16 f32): B[k][n] = W[n][k]; same K-half split by lane group
// C/D (16x16 f32, 8 VGPRs): row = tile_m*16 + j + 8*(lane>=16), col = lane&15
// ---------------------------------------------------------------------------
__global__ void wmma_gemm_bias(const float* __restrict__ A,
                               const float* __restrict__ W,
                               const float* __restrict__ bias,
                               float* __restrict__ C,
                               int M, int Kd, int Nd)
{
    const int wave = (blockIdx.x * blockDim.x + threadIdx.x) >> 5;
    const int lane = threadIdx.x & 31;
    const int ntn  = Nd >> 4;                 // N-tiles
    const int tm   = wave / ntn;              // M-tile
    const int tn   = wave - tm * ntn;         // N-tile
    if (tm >= (M >> 4)) return;               // wave-uniform guard (EXEC stays full)

    const int half  = (lane >> 4);            // 0: K-pair {0,1}, 1: K-pair {2,3}
    const int row   = tm * 16 + (lane & 15);  // A source row for this lane
    const int col   = tn * 16 + (lane & 15);  // B source col (= W row)
    const int koff  = half * 2;

    const float* __restrict__ arow = A + (size_t)row * Kd + koff;
    const float* __restrict__ wrow = W + (size_t)col * Kd + koff;

    v8f acc = {};
    for (int kb = 0; kb < Kd; kb += 4) {
        v2f a, b;
        a.x = arow[kb + 0];
        a.y = arow[kb + 1];
        b.x = wrow[kb + 0];
        b.y = wrow[kb + 1];
        acc = __builtin_amdgcn_wmma_f32_16x16x4_f32(
            /*neg_a=*/false, a, /*neg_b=*/false, b,
            /*c_mod=*/(short)0, acc, /*reuse_a=*/false, /*reuse_b=*/false);
    }

    const float bv = bias[col];
    const int rbase = tm * 16 + (half << 3);
#pragma unroll
    for (int j = 0; j < 8; ++j) {
        C[(size_t)(rbase + j) * Nd + col] = acc[j] + bv;
    }
}

// ---------------------------------------------------------------------------
// Layer-1 message + mean:  h[n,d] = (1/32) * sum_k relu(v[dst[e]][d] + u[src[e]][d]*sp[e])
// One 64-thread block per node (lane = feature dim d, D=64). Edge triples
// staged in LDS once, then 32 perfectly-coalesced 64-wide row reads from the
// L2-resident u/v tables (12.8 MB each << 192 MB L2).
// ---------------------------------------------------------------------------
__global__ void edge_mean_kernel(const float* __restrict__ u,
                                 const float* __restrict__ v,
                                 const float* __restrict__ sp,
                                 const int* __restrict__ src,
                                 const int* __restrict__ dst,
                                 float* __restrict__ h)
{
    __shared__ int   ssrc[32];
    __shared__ int   sdst[32];
    __shared__ float ssp[32];

    const int n = blockIdx.x;
    const int d = threadIdx.x;        // 0..63
    const int ebase = n << 5;         // n*K, anchor_eid == arange(E)

    if (d < 32) {
        ssrc[d] = src[ebase + d];
        ssp[d]  = sp[ebase + d];
    } else {
        sdst[d - 32] = dst[ebase + d - 32];
    }
    __syncthreads();

    float acc = 0.f;
#pragma unroll 4
    for (int k = 0; k < 32; ++k) {
        if (k + 1 < 32) {  // hint next gather rows toward the cache
            __builtin_prefetch(&v[(size_t)sdst[k + 1] * 64], 0, 3);
            __builtin_prefetch(&u[(size_t)ssrc[k + 1] * 64], 0, 3);
        }
        const float m = v[(size_t)sdst[k] * 64 + d]
                      + u[(size_t)ssrc[k] * 64 + d] * ssp[k];
        acc += fmaxf(m, 0.f);
    }
    h[(size_t)n * 64 + d] = acc * (1.0f / 32.0f);
}

// ---------------------------------------------------------------------------
// Layer-2 position head + row L2-normalize.
// One wave32 per node: lane = feature dim (OUT_DIM=32).
// For each k: msg = relu(v2[dst]+u2[src]*sp); p_k = wp2 . msg + bp2 via
// wave32 shfl_xor butterfly; lane k keeps p_k. Then a second butterfly for
// ||p||_2 and the normalized write-out (out[n, lane]).
// ---------------------------------------------------------------------------
__global__ void pos_norm_kernel(const float* __restrict__ u2,
                                const float* __restrict__ v2,
                                const float* __restrict__ sp,
                                const int* __restrict__ src,
                                const int* __restrict__ dst,
                                const float* __restrict__ wp2,
                                const float* __restrict__ bp2,
                                float* __restrict__ out,
                                int N)
{
    const int lane = threadIdx.x & 31;
    const int n = blockIdx.x * (blockDim.x >> 5) + (threadIdx.x >> 5);
    if (n >= N) return;

    const int ebase = n << 5;
    const float w = wp2[lane];
    const float bp = bp2[0];

    float pk = 0.f;
    for (int k = 0; k < 32; ++k) {
        const int e = ebase + k;
        const int s = src[e];          // uniform across wave -> scalarized
        const int t = dst[e];
        const float dd = sp[e];
        const float m = fmaxf(v2[(size_t)t * 32 + lane]
                              + u2[(size_t)s * 32 + lane] * dd, 0.f);
        float c = m * w;
#pragma unroll
        for (int off = 16; off >= 1; off >>= 1)
            c += __shfl_xor(c, off, 32);
        if (lane == k) pk = c + bp;
    }

    float sq = pk * pk;
#pragma unroll
    for (int off = 16; off >= 1; off >>= 1)
        sq += __shfl_xor(sq, off, 32);
    const float nrm = fmaxf(sqrtf(sq), 1e-12f);
    out[ebase + lane] = pk / nrm;
}

// ---------------------------------------------------------------------------
extern "C" void kernel_launch(void* const* d_in, const int* in_sizes, int n_in,
                              void* d_out, int out_size, void* d_ws, size_t ws_size,
                              hipStream_t stream)
{
    const float* feat  = (const float*)d_in[0];   // [N,128]
    const float* dists = (const float*)d_in[1];   // [N,32] -> sp_dist flat [E]
    const int*   src   = (const int*)  d_in[2];   // [E]
    const int*   dst   = (const int*)  d_in[3];   // [E]
    const float* w_pre = (const float*)d_in[4];   // [64,128]
    const float* b_pre = (const float*)d_in[5];   // [64]
    const float* Wu1   = (const float*)d_in[6];   // [64,64]
    const float* bu1   = (const float*)d_in[7];
    const float* Wv1   = (const float*)d_in[8];   // [64,64]
    const float* bv1   = (const float*)d_in[9];
    // d_in[10] (wp1), d_in[11] (bp1): layer-1 position output is discarded
    const float* Wu2   = (const float*)d_in[12];  // [32,64]
    const float* bu2   = (const float*)d_in[13];
    const float* Wv2   = (const float*)d_in[14];  // [32,64]
    const float* bv2   = (const float*)d_in[15];
    const float* wp2   = (const float*)d_in[16];  // [1,32]
    const float* bp2   = (const float*)d_in[17];  // [1]
    float* out = (float*)d_out;                   // [N,32]

    const int N = in_sizes[1] / 32;               // 50000 (divisible by 16)

    // workspace partition (floats): x,u1,v1,h are N*64 ; u2,v2 are N*32  (64 MB total)
    float* x  = (float*)d_ws;
    float* u1 = x  + (size_t)N * 64;
    float* v1 = u1 + (size_t)N * 64;
    float* h  = v1 + (size_t)N * 64;
    float* u2 = h  + (size_t)N * 64;
    float* v2 = u2 + (size_t)N * 32;

    const int mt = N >> 4;                        // 3125 M-tiles

    // x = feat @ w_pre^T + b_pre        (Kd=128, Nd=64)
    {
        int waves = mt * (64 >> 4);
        wmma_gemm_bias<<<dim3((waves + 7) / 8), 256, 0, stream>>>(
            feat, w_pre, b_pre, x, N, 128, 64);
    }
    // u1 = x @ Wu1^T + bu1 ; v1 = x @ Wv1^T + bv1     (Kd=64, Nd=64)
    {
        int waves = mt * (64 >> 4);
        dim3 grid((waves + 7) / 8);
        wmma_gemm_bias<<<grid, 256, 0, stream>>>(x, Wu1, bu1, u1, N, 64, 64);
        wmma_gemm_bias<<<grid, 256, 0, stream>>>(x, Wv1, bv1, v1, N, 64, 64);
    }
    // h = mean_k relu(v1[dst] + u1[src]*sp)
    edge_mean_kernel<<<dim3(N), 64, 0, stream>>>(u1, v1, dists, src, dst, h);

    // u2 = h @ Wu2^T + bu2 ; v2 = h @ Wv2^T + bv2     (Kd=64, Nd=32)
    {
        int waves = mt * (32 >> 4);
        dim3 grid((waves + 7) / 8);
        wmma_gemm_bias<<<grid, 256, 0, stream>>>(h, Wu2, bu2, u2, N, 64, 32);
        wmma_gemm_bias<<<grid, 256, 0, stream>>>(h, Wv2, bv2, v2, N, 64, 32);
    }
    // position head + L2 normalize (8 nodes / 256-thread block)
    pos_norm_kernel<<<dim3((N + 7) / 8), 256, 0, stream>>>(
        u2, v2, dists, src, dst, wp2, bp2, out, N);
}